// IOHMambaPredictor_40286793236741
// MI455X (gfx1250) — compile-verified
//
#include <hip/hip_runtime.h>
#include <hip/hip_bf16.h>
#include <math.h>

// ---------------------------------------------------------------------------
// Model constants (from reference)
// ---------------------------------------------------------------------------
#define Bsz      32
#define Lseq     1024
#define BL       (Bsz * Lseq)       // 32768 rows
#define DMODEL   256
#define DINNER   512
#define DSTATE   16
#define DCONV    4
#define DTRANK   16
#define DBCDIM   (DTRANK + 2 * DSTATE)   // 48
#define LNEPS    1e-5f

typedef __attribute__((ext_vector_type(2))) float v2f;
typedef __attribute__((ext_vector_type(8))) float v8f;

// ---------------------------------------------------------------------------
// Generic fp32 WMMA GEMM:  C[m,n] = sum_k A[m,k] * W[n,k]   (i.e. A @ W.T)
// Each wave computes a 16 x (16*NT) strip, reusing the A fragment NT times.
// 4 waves per block cover 64 rows.  grid = dim3(N/(16*NT), M/64), block = 128.
// Uses V_WMMA_F32_16X16X4_F32; K must be a multiple of 4 (pipelined by 4).
//
// Fragments are software-pipelined one K-step deep: loads for step k+4 are
// issued before the WMMAs of step k, so the loads stay outstanding behind the
// WMMA accumulator chain instead of forcing s_wait_loadcnt 0 per matrix op.
//
// A-frag layout (32-bit A 16x4): lane<16  -> row=lane,    holds K = k0+0, k0+1
//                                lane>=16 -> row=lane-16, holds K = k0+2, k0+3
// B-frag layout (4x16):          lane<16  -> col=lane,    holds K = k0+0, k0+1
//                                lane>=16 -> col=lane-16, holds K = k0+2, k0+3
// C/D layout: VGPR r: lanes 0-15 -> row m0+r, col n0+lane;
//                     lanes 16-31 -> row m0+8+r, col n0+lane-16
// ---------------------------------------------------------------------------
template<int NT>
__global__ void wmma_gemm_f32(const float* __restrict__ A,
                              const float* __restrict__ W,
                              float* __restrict__ C,
                              int K, int lda, int ldw, int ldc)
{
    const int lane = threadIdx.x & 31;
    const int wave = threadIdx.x >> 5;
    const int m0   = (blockIdx.y * 4 + wave) * 16;
    const int n0   = blockIdx.x * (16 * NT);
    const int half = lane >> 4;       // 0 or 1
    const int l16  = lane & 15;

    const float* ap  = A + (size_t)(m0 + l16) * lda + 2 * half;
    const float* bp0 = W + (size_t)(n0 + l16) * ldw + 2 * half;

    v8f acc[NT] = {};

    // prologue: load fragments for k = 0
    v2f a_cur;
    v2f b_cur[NT];
    a_cur.x = ap[0];
    a_cur.y = ap[1];
#pragma unroll
    for (int nt = 0; nt < NT; ++nt) {
        const float* bp = bp0 + (size_t)(nt * 16) * ldw;
        b_cur[nt].x = bp[0];
        b_cur[nt].y = bp[1];
    }

    for (int k = 4; k < K; k += 4) {
        // issue next step's loads first (kept outstanding behind the WMMAs)
        v2f a_nxt;
        v2f b_nxt[NT];
        a_nxt.x = ap[k];
        a_nxt.y = ap[k + 1];
#pragma unroll
        for (int nt = 0; nt < NT; ++nt) {
            const float* bp = bp0 + (size_t)(nt * 16) * ldw;
            b_nxt[nt].x = bp[k];
            b_nxt[nt].y = bp[k + 1];
        }
        // matrix ops on the current fragments
#pragma unroll
        for (int nt = 0; nt < NT; ++nt)
            acc[nt] = __builtin_amdgcn_wmma_f32_16x16x4_f32(
                false, a_cur, false, b_cur[nt],
                (short)0, acc[nt], false, false);
        a_cur = a_nxt;
#pragma unroll
        for (int nt = 0; nt < NT; ++nt)
            b_cur[nt] = b_nxt[nt];
    }

    // epilogue: last K-step
#pragma unroll
    for (int nt = 0; nt < NT; ++nt)
        acc[nt] = __builtin_amdgcn_wmma_f32_16x16x4_f32(
            false, a_cur, false, b_cur[nt],
            (short)0, acc[nt], false, false);

#pragma unroll
    for (int nt = 0; nt < NT; ++nt) {
        float* cp = C + (size_t)(m0 + half * 8) * ldc + n0 + nt * 16 + l16;
#pragma unroll
        for (int r = 0; r < 8; ++r)
            cp[(size_t)r * ldc] = acc[nt][r];
    }
}

// ---------------------------------------------------------------------------
// proj1 (K=4) + bias + LayerNorm.  grid = BL, block = 256 (one row / block).
// ---------------------------------------------------------------------------
__global__ void proj1_ln_kernel(const float* __restrict__ xseq,
                                const float* __restrict__ w,     // (256,4)
                                const float* __restrict__ bias,
                                const float* __restrict__ g,
                                const float* __restrict__ be,
                                float* __restrict__ out)
{
    const int row = blockIdx.x;
    const int j   = threadIdx.x;
    const float* xr = xseq + (size_t)row * 4;
    const float* wr = w + (size_t)j * 4;
    float v = bias[j] + xr[0] * wr[0] + xr[1] * wr[1] + xr[2] * wr[2] + xr[3] * wr[3];

    __shared__ float sd[256];
    sd[j] = v;
    __syncthreads();
    for (int s = 128; s > 0; s >>= 1) { if (j < s) sd[j] += sd[j + s]; __syncthreads(); }
    const float mu = sd[0] * (1.0f / 256.0f);
    __syncthreads();
    const float d = v - mu;
    sd[j] = d * d;
    __syncthreads();
    for (int s = 128; s > 0; s >>= 1) { if (j < s) sd[j] += sd[j + s]; __syncthreads(); }
    const float var = sd[0] * (1.0f / 256.0f);
    out[(size_t)row * 256 + j] = d * rsqrtf(var + LNEPS) * g[j] + be[j];
}

// ---------------------------------------------------------------------------
// bias + LayerNorm on a precomputed (BL,256) GEMM result.
// ---------------------------------------------------------------------------
__global__ void bias_ln_kernel(const float* __restrict__ in,
                               const float* __restrict__ bias,
                               const float* __restrict__ g,
                               const float* __restrict__ be,
                               float* __restrict__ out)
{
    const int row = blockIdx.x;
    const int j   = threadIdx.x;
    float v = in[(size_t)row * 256 + j] + bias[j];

    __shared__ float sd[256];
    sd[j] = v;
    __syncthreads();
    for (int s = 128; s > 0; s >>= 1) { if (j < s) sd[j] += sd[j + s]; __syncthreads(); }
    const float mu = sd[0] * (1.0f / 256.0f);
    __syncthreads();
    const float d = v - mu;
    sd[j] = d * d;
    __syncthreads();
    for (int s = 128; s > 0; s >>= 1) { if (j < s) sd[j] += sd[j + s]; __syncthreads(); }
    const float var = sd[0] * (1.0f / 256.0f);
    out[(size_t)row * 256 + j] = d * rsqrtf(var + LNEPS) * g[j] + be[j];
}

// ---------------------------------------------------------------------------
// Depthwise causal conv1d (k=4) + SiLU on xi = xz[:, :512].
// grid = dim3(Lseq, Bsz), block = 512 (one channel / thread).
// ---------------------------------------------------------------------------
__global__ void conv_silu_kernel(const float* __restrict__ xz,     // (BL,1024)
                                 const float* __restrict__ conv_w, // (512,4)
                                 const float* __restrict__ conv_b,
                                 float* __restrict__ xc)           // (BL,512)
{
    const int l = blockIdx.x;
    const int b = blockIdx.y;
    const int c = threadIdx.x;
    float acc = conv_b[c];
#pragma unroll
    for (int k = 0; k < DCONV; ++k) {
        const int lt = l + k - (DCONV - 1);
        if (lt >= 0)
            acc += conv_w[c * DCONV + k] * xz[((size_t)(b * Lseq + lt)) * 1024 + c];
    }
    const float sig = 1.0f / (1.0f + expf(-acc));
    xc[((size_t)(b * Lseq + l)) * DINNER + c] = acc * sig;
}

// ---------------------------------------------------------------------------
// softplus(x + bias) over (BL,512); bias per channel.
// ---------------------------------------------------------------------------
__global__ void softplus_bias_kernel(float* __restrict__ dtb,
                                     const float* __restrict__ bias)
{
    const size_t i = (size_t)blockIdx.x * blockDim.x + threadIdx.x;
    const int c = (int)(i & (DINNER - 1));
    const float v = dtb[i] + bias[c];
    dtb[i] = (v > 20.0f) ? v : log1pf(expf(v));
}

// ---------------------------------------------------------------------------
// Selective scan + skip (x*D) + z-gating.  grid = Bsz, block = 512.
// Each thread owns one channel d; 16 states + 16 A values in registers.
// B_t / C_t (48-float dbc row) staged into LDS per timestep using CDNA5
// GLOBAL_LOAD_ASYNC_TO_LDS_B128 (12 lanes x 16B), drained via s_wait_asynccnt.
// ---------------------------------------------------------------------------
__global__ void scan_kernel(const float* __restrict__ dt,    // (BL,512)
                            const float* __restrict__ dbc,   // (BL,48)
                            const float* __restrict__ xc,    // (BL,512)
                            const float* __restrict__ xz,    // (BL,1024), z in cols 512..1023
                            const float* __restrict__ A_log, // (512,16)
                            const float* __restrict__ Dp,    // (512)
                            float* __restrict__ y)           // (BL,512)
{
    const int b = blockIdx.x;
    const int d = threadIdx.x;

    float Arow[DSTATE];
#pragma unroll
    for (int s = 0; s < DSTATE; ++s)
        Arow[s] = -expf(A_log[d * DSTATE + s]);
    float h[DSTATE];
#pragma unroll
    for (int s = 0; s < DSTATE; ++s) h[s] = 0.0f;
    const float Dv = Dp[d];

    __shared__ float stage[DBCDIM];          // [0..15]=dt part (unused), [16..31]=B, [32..47]=C
    const unsigned lds_base = (unsigned)(size_t)(&stage[0]);
    const unsigned my_lds   = lds_base + (unsigned)(d * 16);   // lane d copies 16 bytes

    for (int t = 0; t < Lseq; ++t) {
        const size_t row = (size_t)(b * Lseq + t);

        if (d < 32) {
            if (d < 12) {
                // async-copy one 16-byte chunk of the dbc row straight into LDS
                const float* gsrc = dbc + row * DBCDIM + d * 4;
                asm volatile("global_load_async_to_lds_b128 %0, %1, off"
                             :: "v"(my_lds), "v"(gsrc)
                             : "memory");
            }
            // wave 0 drains its async copies before the workgroup barrier
            asm volatile("s_wait_asynccnt 0" ::: "memory");
        }
        if (t + 1 < Lseq)
            __builtin_prefetch(dbc + (row + 1) * DBCDIM, 0, 0);  // global_prefetch_b8
        __syncthreads();

        const float dtv = dt[row * DINNER + d];
        const float xv  = xc[row * DINNER + d];
        const float zv  = xz[row * 1024 + DINNER + d];
        const float du  = dtv * xv;

        float yacc = 0.0f;
#pragma unroll
        for (int s = 0; s < DSTATE; ++s) {
            h[s] = expf(dtv * Arow[s]) * h[s] + du * stage[DTRANK + s];
            yacc += h[s] * stage[DTRANK + DSTATE + s];
        }
        const float yv  = yacc + xv * Dv;
        const float sig = 1.0f / (1.0f + expf(-zv));
        y[row * DINNER + d] = yv * (zv * sig);
        __syncthreads();
    }
}

// ---------------------------------------------------------------------------
// Mean over L:  pooled[b,c] = mean_t mo[b,t,c].  grid = Bsz, block = 256.
// ---------------------------------------------------------------------------
__global__ void mean_pool_kernel(const float* __restrict__ mo,
                                 float* __restrict__ pooled)
{
    const int b = blockIdx.x;
    const int c = threadIdx.x;
    const float* p = mo + (size_t)b * Lseq * 256 + c;
    float s = 0.0f;
    for (int t = 0; t < Lseq; ++t) s += p[(size_t)t * 256];
    pooled[b * 256 + c] = s * (1.0f / (float)Lseq);
}

// ---------------------------------------------------------------------------
// Head: h1 = elu(concat(pooled, static) @ h1_w.T + h1_b); out = h1 @ h2_w.T + h2_b
// grid = Bsz, block = 128.
// ---------------------------------------------------------------------------
__global__ void head_kernel(const float* __restrict__ pooled,
                            const float* __restrict__ xstatic,  // (Bsz,5)
                            const float* __restrict__ h1w,      // (128,261)
                            const float* __restrict__ h1b,
                            const float* __restrict__ h2w,      // (1,128)
                            const float* __restrict__ h2b,
                            float* __restrict__ out)
{
    const int b = blockIdx.x;
    const int j = threadIdx.x;
    const float* wr = h1w + (size_t)j * 261;
    const float* pr = pooled + b * 256;
    float acc = h1b[j];
    for (int k = 0; k < 256; ++k) acc += wr[k] * pr[k];
#pragma unroll
    for (int k = 0; k < 5; ++k) acc += wr[256 + k] * xstatic[b * 5 + k];
    const float a = (acc > 0.0f) ? acc : (expf(acc) - 1.0f);  // ELU(alpha=1)

    __shared__ float sd[128];
    sd[j] = a * h2w[j];
    __syncthreads();
    for (int s = 64; s > 0; s >>= 1) { if (j < s) sd[j] += sd[j + s]; __syncthreads(); }
    if (j == 0) out[b] = sd[0] + h2b[0];
}

// ---------------------------------------------------------------------------
// Host-side orchestration of one Mamba block.
// ---------------------------------------------------------------------------
static void run_mamba(const float* x,        // (BL,256) input
                      const float* in_w,     // (1024,256)
                      const float* conv_w, const float* conv_b,
                      const float* xproj_w,  // (48,512)
                      const float* dt_w,     // (512,16)
                      const float* dt_b,
                      const float* A_log, const float* Dp,
                      const float* out_w,    // (256,512)
                      float* xz, float* xc, float* dbc, float* dtb,
                      float* ybuf, float* mo,
                      hipStream_t stream)
{
    // in-projection: (BL,256) @ (1024,256).T -> xz (BL,1024)
    wmma_gemm_f32<4><<<dim3(1024 / 64, BL / 64), 128, 0, stream>>>(
        x, in_w, xz, /*K=*/DMODEL, /*lda=*/DMODEL, /*ldw=*/DMODEL, /*ldc=*/1024);

    // depthwise causal conv + SiLU on xi -> xc (BL,512)
    conv_silu_kernel<<<dim3(Lseq, Bsz), DINNER, 0, stream>>>(xz, conv_w, conv_b, xc);

    // x-projection: (BL,512) @ (48,512).T -> dbc (BL,48)
    wmma_gemm_f32<3><<<dim3(1, BL / 64), 128, 0, stream>>>(
        xc, xproj_w, dbc, /*K=*/DINNER, /*lda=*/DINNER, /*ldw=*/DINNER, /*ldc=*/DBCDIM);

    // dt projection: dbc[:, :16] @ (512,16).T -> dtb (BL,512)
    wmma_gemm_f32<4><<<dim3(DINNER / 64, BL / 64), 128, 0, stream>>>(
        dbc, dt_w, dtb, /*K=*/DTRANK, /*lda=*/DBCDIM, /*ldw=*/DTRANK, /*ldc=*/DINNER);

    // softplus(dt + dt_b)
    softplus_bias_kernel<<<(BL * DINNER) / 256, 256, 0, stream>>>(dtb, dt_b);

    // selective scan + skip + z-gate -> ybuf (BL,512)
    scan_kernel<<<Bsz, DINNER, 0, stream>>>(dtb, dbc, xc, xz, A_log, Dp, ybuf);

    // out-projection: (BL,512) @ (256,512).T -> mo (BL,256)
    wmma_gemm_f32<4><<<dim3(DMODEL / 64, BL / 64), 128, 0, stream>>>(
        ybuf, out_w, mo, /*K=*/DINNER, /*lda=*/DINNER, /*ldw=*/DINNER, /*ldc=*/DMODEL);
}

// ---------------------------------------------------------------------------
// Entry point.
// ---------------------------------------------------------------------------
extern "C" void kernel_launch(void* const* d_in, const int* in_sizes, int n_in,
                              void* d_out, int out_size, void* d_ws, size_t ws_size,
                              hipStream_t stream)
{
    (void)in_sizes; (void)n_in; (void)out_size; (void)ws_size;

    const float* x_seq    = (const float*)d_in[0];
    const float* x_static = (const float*)d_in[1];
    const float* p1_w  = (const float*)d_in[2];
    const float* p1_b  = (const float*)d_in[3];
    const float* ln1_g = (const float*)d_in[4];
    const float* ln1_b = (const float*)d_in[5];
    const float* p2_w  = (const float*)d_in[6];
    const float* p2_b  = (const float*)d_in[7];
    const float* ln2_g = (const float*)d_in[8];
    const float* ln2_b = (const float*)d_in[9];
    const float* h1_w  = (const float*)d_in[10];
    const float* h1_b  = (const float*)d_in[11];
    const float* h2_w  = (const float*)d_in[12];
    const float* h2_b  = (const float*)d_in[13];

    const float* m1p[9]; for (int i = 0; i < 9; ++i) m1p[i] = (const float*)d_in[14 + i];
    const float* m2p[9]; for (int i = 0; i < 9; ++i) m2p[i] = (const float*)d_in[23 + i];
    // order: in_w, conv_w, conv_b, xproj_w, dt_w, dt_b, A_log, D, out_w

    // ---- workspace layout (floats) ----
    float* ws   = (float*)d_ws;
    float* xln  = ws;                           // BL*256 (mamba input)
    float* xz   = xln + (size_t)BL * 256;       // BL*1024
    float* xc   = xz  + (size_t)BL * 1024;      // BL*512
    float* dbc  = xc  + (size_t)BL * 512;       // BL*48
    float* dtb  = dbc + (size_t)BL * DBCDIM;    // BL*512 (also reused as GEMM tmp)
    float* ybuf = dtb + (size_t)BL * 512;       // BL*512
    float* mo   = ybuf + (size_t)BL * 512;      // BL*256 (mamba output)
    float* pooled = mo + (size_t)BL * 256;      // Bsz*256

    // 1) input projection (K=4) + LN1
    proj1_ln_kernel<<<BL, 256, 0, stream>>>(x_seq, p1_w, p1_b, ln1_g, ln1_b, xln);

    // 2) mamba block 1
    run_mamba(xln, m1p[0], m1p[1], m1p[2], m1p[3], m1p[4], m1p[5], m1p[6], m1p[7], m1p[8],
              xz, xc, dbc, dtb, ybuf, mo, stream);

    // 3) proj2 (256->256) + LN2   (dtb reused as BL*256 temp; free after scan)
    wmma_gemm_f32<4><<<dim3(DMODEL / 64, BL / 64), 128, 0, stream>>>(
        mo, p2_w, dtb, /*K=*/DMODEL, /*lda=*/DMODEL, /*ldw=*/DMODEL, /*ldc=*/DMODEL);
    bias_ln_kernel<<<BL, 256, 0, stream>>>(dtb, p2_b, ln2_g, ln2_b, xln);

    // 4) mamba block 2
    run_mamba(xln, m2p[0], m2p[1], m2p[2], m2p[3], m2p[4], m2p[5], m2p[6], m2p[7], m2p[8],
              xz, xc, dbc, dtb, ybuf, mo, stream);

    // 5) mean pool over L, then MLP head
    mean_pool_kernel<<<Bsz, 256, 0, stream>>>(mo, pooled);
    head_kernel<<<Bsz, 128, 0, stream>>>(pooled, x_static, h1_w, h1_b, h2_w, h2_b,
                                         (float*)d_out);
}